// GATConv_40767829574219
// MI455X (gfx1250) — compile-verified
//
#include <hip/hip_runtime.h>

#define IN_DIM   128
#define HD       64
#define H_HEADS  4
#define D_DIM    16
#define NEG_SLOPE 0.2f

typedef __attribute__((ext_vector_type(2))) float v2f;
typedef __attribute__((ext_vector_type(8))) float v8f;

// Order-preserving float<->uint encoding so segment-max can use native u32 max atomics.
__device__ __forceinline__ unsigned fenc(float f) {
    unsigned u = __float_as_uint(f);
    return (u & 0x80000000u) ? ~u : (u | 0x80000000u);
}
__device__ __forceinline__ float fdec(unsigned u) {
    return (u & 0x80000000u) ? __uint_as_float(u ^ 0x80000000u) : __uint_as_float(~u);
}
__device__ __forceinline__ float lrelu(float x) {
    return x > 0.0f ? x : NEG_SLOPE * x;
}

// ---------------------------------------------------------------------------
// Kernel 1: init out = bias (broadcast), m = enc(-1e30), s = 0
// ---------------------------------------------------------------------------
__global__ void gat_init(float* __restrict__ out, float* __restrict__ s_sum,
                         unsigned* __restrict__ m_enc, const float* __restrict__ bias,
                         int N) {
    int idx = blockIdx.x * blockDim.x + threadIdx.x;
    if (idx < N * HD) out[idx] = bias[idx & (HD - 1)];
    if (idx < N * H_HEADS) {
        s_sum[idx] = 0.0f;
        m_enc[idx] = fenc(-1e30f);
    }
}

// ---------------------------------------------------------------------------
// Kernel 2: feat = h @ W^T via V_WMMA_F32_16X16X4_F32 (f32 tensor op, no
// precision downgrade). Block = 128 threads (4 waves). Wave w computes the
// 16x16 tile feat[row0:row0+16, 16w:16w+16] (= head w), chaining 32 WMMAs
// over K=128.
//   A frag (16x4 f32): lane l<16 -> M=l, K={k0,k0+1}; lane>=16 -> K={k0+2,k0+3}
//   B frag mirrored; B[k][n] = W[n][k] (W row-major) -> contiguous float2 load
//   C/D: vgpr i -> row (i + 8*half), col lane%16
// ---------------------------------------------------------------------------
__global__ void __launch_bounds__(128)
gat_gemm(const float* __restrict__ hmat, const float* __restrict__ W,
         float* __restrict__ feat, int N) {
    const int row0 = blockIdx.x * 16;
    const int wave = threadIdx.x >> 5;     // 0..3 == head
    const int lane = threadIdx.x & 31;
    const int half = lane >> 4;            // 0 or 1
    const int l16  = lane & 15;

    int arow = row0 + l16;
    if (arow >= N) arow = N - 1;           // clamp (keeps EXEC all-1s)
    const int ncol = wave * 16 + l16;

    const float* ap = hmat + (size_t)arow * IN_DIM + 2 * half;
    const float* bp = W    + (size_t)ncol * IN_DIM + 2 * half;

    v8f c = {};
#pragma unroll
    for (int k0 = 0; k0 < IN_DIM; k0 += 4) {
        float2 av = *(const float2*)(ap + k0);
        float2 bv = *(const float2*)(bp + k0);
        v2f a; a.x = av.x; a.y = av.y;
        v2f b; b.x = bv.x; b.y = bv.y;
        // (neg_a, A, neg_b, B, c_mod, C, reuse_a, reuse_b)
        c = __builtin_amdgcn_wmma_f32_16x16x4_f32(
                false, a, false, b, (short)0, c, false, false);
    }

#pragma unroll
    for (int i = 0; i < 8; ++i) {
        const int r = row0 + i + 8 * half;
        if (r < N) feat[(size_t)r * HD + wave * 16 + l16] = c[i];
    }
}

// ---------------------------------------------------------------------------
// Kernel 3: el/er[n,h] = <feat[n,h,:], attn_l/r[h,:]>
// ---------------------------------------------------------------------------
__global__ void gat_logits(const float* __restrict__ feat,
                           const float* __restrict__ attn_l,
                           const float* __restrict__ attn_r,
                           float* __restrict__ el, float* __restrict__ er, int N) {
    int idx = blockIdx.x * blockDim.x + threadIdx.x;
    if (idx >= N * H_HEADS) return;
    const int n  = idx >> 2;
    const int hh = idx & 3;
    const float* f  = feat + (size_t)n * HD + hh * D_DIM;
    const float* al = attn_l + hh * D_DIM;
    const float* ar = attn_r + hh * D_DIM;
    float sl = 0.0f, sr = 0.0f;
#pragma unroll
    for (int d = 0; d < D_DIM; ++d) {
        float v = f[d];
        sl = fmaf(v, al[d], sl);
        sr = fmaf(v, ar[d], sr);
    }
    el[idx] = sl;
    er[idx] = sr;
}

// ---------------------------------------------------------------------------
// Kernel 4: segment max via native u32 max atomic (L2-resident)
// ---------------------------------------------------------------------------
__global__ void gat_edge_max(const int* __restrict__ src, const int* __restrict__ dst,
                             const float* __restrict__ el, const float* __restrict__ er,
                             unsigned* __restrict__ m_enc, int E) {
    int idx = blockIdx.x * blockDim.x + threadIdx.x;
    if (idx >= E * H_HEADS) return;
    const int e  = idx >> 2;
    const int hh = idx & 3;
    const int s = src[e], d = dst[e];
    const float sc = lrelu(el[s * H_HEADS + hh] + er[d * H_HEADS + hh]);
    atomicMax(&m_enc[d * H_HEADS + hh], fenc(sc));
}

// ---------------------------------------------------------------------------
// Kernel 5: segment sum of exp(e - m) via native f32 add atomic
// ---------------------------------------------------------------------------
__global__ void gat_edge_sum(const int* __restrict__ src, const int* __restrict__ dst,
                             const float* __restrict__ el, const float* __restrict__ er,
                             const unsigned* __restrict__ m_enc,
                             float* __restrict__ s_sum, int E) {
    int idx = blockIdx.x * blockDim.x + threadIdx.x;
    if (idx >= E * H_HEADS) return;
    const int e  = idx >> 2;
    const int hh = idx & 3;
    const int s = src[e], d = dst[e];
    const int di = d * H_HEADS + hh;
    const float sc = lrelu(el[s * H_HEADS + hh] + er[di]);
    atomicAdd(&s_sum[di], __expf(sc - fdec(m_enc[di])));
}

// ---------------------------------------------------------------------------
// Kernel 6: out[dst] += feat[src] * alpha  — one thread per (edge, k<64).
// feat reads and out atomics are contiguous 256B per edge (L2-friendly).
// ---------------------------------------------------------------------------
__global__ void gat_aggregate(const int* __restrict__ src, const int* __restrict__ dst,
                              const float* __restrict__ el, const float* __restrict__ er,
                              const unsigned* __restrict__ m_enc,
                              const float* __restrict__ s_sum,
                              const float* __restrict__ feat,
                              float* __restrict__ out, int E) {
    int idx = blockIdx.x * blockDim.x + threadIdx.x;
    if (idx >= E * HD) return;
    const int e  = idx >> 6;
    const int k  = idx & 63;
    const int hh = k >> 4;
    const int s = src[e], d = dst[e];
    const int di = d * H_HEADS + hh;
    const float sc = lrelu(el[s * H_HEADS + hh] + er[di]);
    const float ss = s_sum[di];
    const float alpha = __expf(sc - fdec(m_enc[di])) / (ss > 0.0f ? ss : 1.0f);
    atomicAdd(&out[(size_t)d * HD + k], feat[(size_t)s * HD + k] * alpha);
}

// ---------------------------------------------------------------------------
extern "C" void kernel_launch(void* const* d_in, const int* in_sizes, int n_in,
                              void* d_out, int out_size, void* d_ws, size_t ws_size,
                              hipStream_t stream) {
    const float* hmat   = (const float*)d_in[0];
    const float* W      = (const float*)d_in[1];
    const float* attn_l = (const float*)d_in[2];
    const float* attn_r = (const float*)d_in[3];
    const float* bias   = (const float*)d_in[4];
    const int*   src    = (const int*)d_in[5];
    const int*   dst    = (const int*)d_in[6];

    const int N = in_sizes[0] / IN_DIM;   // 100000
    const int E = in_sizes[5];            // 3200000

    float* out = (float*)d_out;

    // workspace layout (all f32/u32, ~32 MB total)
    float*    feat  = (float*)d_ws;
    float*    el    = feat  + (size_t)N * HD;
    float*    er    = el    + (size_t)N * H_HEADS;
    float*    s_sum = er    + (size_t)N * H_HEADS;
    unsigned* m_enc = (unsigned*)(s_sum + (size_t)N * H_HEADS);

    const int BS = 256;
    gat_init<<<(N * HD + BS - 1) / BS, BS, 0, stream>>>(out, s_sum, m_enc, bias, N);
    gat_gemm<<<(N + 15) / 16, 128, 0, stream>>>(hmat, W, feat, N);
    gat_logits<<<(N * H_HEADS + BS - 1) / BS, BS, 0, stream>>>(feat, attn_l, attn_r, el, er, N);
    gat_edge_max<<<((size_t)E * H_HEADS + BS - 1) / BS, BS, 0, stream>>>(src, dst, el, er, m_enc, E);
    gat_edge_sum<<<((size_t)E * H_HEADS + BS - 1) / BS, BS, 0, stream>>>(src, dst, el, er, m_enc, s_sum, E);
    gat_aggregate<<<((size_t)E * HD + BS - 1) / BS, BS, 0, stream>>>(src, dst, el, er, m_enc, s_sum, feat, out, E);
}